// BidirectionalRecurrentModelCore_87153476370860
// MI455X (gfx1250) — compile-verified
//
#include <hip/hip_runtime.h>
#include <hip/hip_bf16.h>
#include <math.h>

// Problem constants (match reference)
#define S_  64
#define B_  256
#define E_  131072
#define H_  256
#define T_  (S_ * B_)          // 16384 nodes
#define NE_ 16
#define DEPTH_ 3

typedef float v2f __attribute__((ext_vector_type(2)));
typedef float v8f __attribute__((ext_vector_type(8)));
typedef unsigned int v4u __attribute__((ext_vector_type(4)));
typedef int v4i __attribute__((ext_vector_type(4)));
typedef int v8i __attribute__((ext_vector_type(8)));

#if defined(__has_builtin)
#  if __has_builtin(__builtin_amdgcn_tensor_load_to_lds) && __has_builtin(__builtin_amdgcn_s_wait_tensorcnt)
#    define HAVE_TDM 1
#  else
#    define HAVE_TDM 0
#  endif
#else
#  define HAVE_TDM 0
#endif

__device__ __forceinline__ float sigmoid_(float x) {
    return 1.0f / (1.0f + __expf(-x));
}

#if HAVE_TDM
// ---------------------------------------------------------------------------
// Tensor Data Mover: async-load a 2D tile (rows x 32 floats) of W into LDS.
//   global tile start = gsrc, row stride = ld_elems floats.
//   pad_enable: insert 4 dwords after every 32 dwords -> LDS row stride 36
//   floats (16B-aligned rows, bank-conflict-free for the WMMA fragment reads).
// Issued by one wave per workgroup; tracked with TENSORcnt.
// 6-arg builtin form (clang-23 / therock-10.0 headers).
// ---------------------------------------------------------------------------
__device__ __forceinline__ void tdm_load_tile(const float* gsrc, unsigned lds_byte_off,
                                              int ld_elems, int rows)
{
    unsigned long long ga = (unsigned long long)(uintptr_t)gsrc;

    v4u g0;
    g0.x = 1u;                                              // count=1 (valid user D#)
    g0.y = lds_byte_off;                                    // lds_addr
    g0.z = (unsigned)(ga & 0xFFFFFFFFu);                    // global_addr[31:0]
    g0.w = (unsigned)((ga >> 32) & 0x01FFFFFFu) | (2u << 30); // global_addr[56:32], type=2

    v8i g1;
    g1[0] = (2 << 16)      // data_size = 4 bytes
          | (1 << 20)      // pad_enable
          | (4 << 22)      // pad_interval: every 32 dwords
          | (3 << 25);     // pad_amount: 4 dwords
    g1[1] = (int)(((unsigned)ld_elems & 0xFFFFu) << 16);    // tensor_dim0[15:0]
    g1[2] = (ld_elems >> 16) & 0xFFFF;                      // tensor_dim0[31:16]; tensor_dim1 lo16 = 0
    g1[3] = 0x10 | (32 << 16);                              // tensor_dim1 = 1M rows; tile_dim0 = 32
    g1[4] = rows;                                           // tile_dim1 = rows; tile_dim2 = 0 (2D)
    g1[5] = ld_elems;                                       // tensor_dim0_stride[31:0]
    g1[6] = 0;                                              // stride hi / dim1_stride lo
    g1[7] = 0;

    v4i gz4 = {0, 0, 0, 0};
    v8i gz8 = {0, 0, 0, 0, 0, 0, 0, 0};
    __builtin_amdgcn_tensor_load_to_lds(g0, g1, gz4, gz4, gz8, 0);
}
#endif

// ---------------------------------------------------------------------------
// Generic WMMA f32 GEMM:  C[M,N] = A[sel(m),:K] @ W[N,K]^T + bias[N]
// Block: 256 threads (8 waves), tile 64(M) x 128(N); wave = 16x64 (4 accums).
// A tile staged in LDS; B tile double-buffered in LDS via TDM (tensor_load_to_lds
// + s_wait_tensorcnt), so the inner loop is pure ds_load + v_wmma.
// Requires M%64==0, N%128==0, K%32==0 (true for every GEMM in this model).
// ---------------------------------------------------------------------------
__global__ __launch_bounds__(256) void gemm_bias_kernel(
    const float* __restrict__ A, int lda, const int* __restrict__ rowidx,
    const float* __restrict__ W, int ldw, const float* __restrict__ bias,
    float* __restrict__ C, int ldc, int K)
{
    __shared__ float a_tile[64][36];
    __shared__ float b_tile[2][128][36];

    const int tid    = threadIdx.x;
    const int lane   = tid & 31;
    const int wid    = tid >> 5;
    const int wave_m = wid >> 1;               // 0..3
    const int wave_n = wid & 1;                // 0..1
    const int m0     = blockIdx.x * 64;
    const int n0     = blockIdx.y * 128;
    const int mlane  = lane & 15;
    const int koff   = (lane >> 4) << 1;       // 0 or 2

    v8f acc[4] = {};

    // cooperative A loader: thread -> (row 0..63, two quads)
    const int ldrow = tid >> 2;
    const int ldq   = (tid & 3) << 2;
    int arow = m0 + ldrow;
    if (rowidx) arow = rowidx[arow];
    const float* aptr = A + (size_t)arow * lda;

    const float* wblk = W + (size_t)n0 * ldw;
    const int nch = K >> 5;
    int buf = 0;

#if HAVE_TDM
    if (wid == 0)
        tdm_load_tile(wblk, (unsigned)(uintptr_t)&b_tile[0][0][0], ldw, 128);
#endif

    for (int c = 0; c < nch; ++c) {
        const int k0 = c << 5;
        __syncthreads();   // previous chunk's compute done: safe to overwrite LDS

        *(float4*)&a_tile[ldrow][ldq]      = *(const float4*)(aptr + k0 + ldq);
        *(float4*)&a_tile[ldrow][ldq + 16] = *(const float4*)(aptr + k0 + ldq + 16);

#if HAVE_TDM
        if (wid == 0) {
            if (c + 1 < nch) {
                tdm_load_tile(wblk + k0 + 32,
                              (unsigned)(uintptr_t)&b_tile[buf ^ 1][0][0], ldw, 128);
                __builtin_amdgcn_s_wait_tensorcnt(1);
            } else {
                __builtin_amdgcn_s_wait_tensorcnt(0);
            }
        }
#else
        {   // fallback: cooperative global->LDS load of the B tile
            const int brow = tid >> 1;
            const int bh   = (tid & 1) << 4;
            const float* bp = wblk + (size_t)brow * ldw + k0 + bh;
            float4* dp = (float4*)&b_tile[buf][brow][bh];
            dp[0] = ((const float4*)bp)[0];
            dp[1] = ((const float4*)bp)[1];
            dp[2] = ((const float4*)bp)[2];
            dp[3] = ((const float4*)bp)[3];
        }
#endif
        __syncthreads();   // A staged + B(buf) resident

#pragma unroll
        for (int kk = 0; kk < 32; kk += 4) {
            v2f a = *(const v2f*)&a_tile[wave_m * 16 + mlane][kk + koff];
#pragma unroll
            for (int nt = 0; nt < 4; ++nt) {
                v2f b = *(const v2f*)&b_tile[buf][wave_n * 64 + nt * 16 + mlane][kk + koff];
                acc[nt] = __builtin_amdgcn_wmma_f32_16x16x4_f32(
                    false, a, false, b, (short)0, acc[nt], false, false);
            }
        }
        buf ^= 1;
    }

#pragma unroll
    for (int nt = 0; nt < 4; ++nt) {
        int col  = n0 + wave_n * 64 + nt * 16 + mlane;
        float bv = bias ? bias[col] : 0.0f;
#pragma unroll
        for (int r = 0; r < 8; ++r) {
            int row = m0 + wave_m * 16 + r + ((lane >> 4) << 3);
            C[(size_t)row * ldc + col] = acc[nt][r] + bv;
        }
    }
}

// ---------------------------------------------------------------------------
// Edge-message GEMM + scatter:  for each edge e:
//   msg[e,:] = node_h[src[e],:] @ Wx^T + bias_table[edge_types[e],:]
//   agg[dst[e],:] += msg[e,:]        (global_atomic_add_f32)
// Same 64x128 tiling; N = 256 -> grid.y = 2; K = 256.
// ---------------------------------------------------------------------------
__global__ __launch_bounds__(256) void msg_gemm_scatter_kernel(
    const float* __restrict__ node_h, const int* __restrict__ src,
    const float* __restrict__ Wx, int ldw,
    const float* __restrict__ bias_table,
    const int* __restrict__ edge_types, const int* __restrict__ dst,
    float* __restrict__ agg, int K)
{
    __shared__ float a_tile[64][36];
    __shared__ float b_tile[2][128][36];

    const int tid    = threadIdx.x;
    const int lane   = tid & 31;
    const int wid    = tid >> 5;
    const int wave_m = wid >> 1;
    const int wave_n = wid & 1;
    const int m0     = blockIdx.x * 64;
    const int n0     = blockIdx.y * 128;
    const int mlane  = lane & 15;
    const int koff   = (lane >> 4) << 1;

    v8f acc[4] = {};

    const int ldrow = tid >> 2;
    const int ldq   = (tid & 3) << 2;
    const int arow  = src[m0 + ldrow];
    const float* aptr = node_h + (size_t)arow * 256;

    const float* wblk = Wx + (size_t)n0 * ldw;
    const int nch = K >> 5;
    int buf = 0;

#if HAVE_TDM
    if (wid == 0)
        tdm_load_tile(wblk, (unsigned)(uintptr_t)&b_tile[0][0][0], ldw, 128);
#endif

    for (int c = 0; c < nch; ++c) {
        const int k0 = c << 5;
        __syncthreads();

        *(float4*)&a_tile[ldrow][ldq]      = *(const float4*)(aptr + k0 + ldq);
        *(float4*)&a_tile[ldrow][ldq + 16] = *(const float4*)(aptr + k0 + ldq + 16);

#if HAVE_TDM
        if (wid == 0) {
            if (c + 1 < nch) {
                tdm_load_tile(wblk + k0 + 32,
                              (unsigned)(uintptr_t)&b_tile[buf ^ 1][0][0], ldw, 128);
                __builtin_amdgcn_s_wait_tensorcnt(1);
            } else {
                __builtin_amdgcn_s_wait_tensorcnt(0);
            }
        }
#else
        {
            const int brow = tid >> 1;
            const int bh   = (tid & 1) << 4;
            const float* bp = wblk + (size_t)brow * ldw + k0 + bh;
            float4* dp = (float4*)&b_tile[buf][brow][bh];
            dp[0] = ((const float4*)bp)[0];
            dp[1] = ((const float4*)bp)[1];
            dp[2] = ((const float4*)bp)[2];
            dp[3] = ((const float4*)bp)[3];
        }
#endif
        __syncthreads();

#pragma unroll
        for (int kk = 0; kk < 32; kk += 4) {
            v2f a = *(const v2f*)&a_tile[wave_m * 16 + mlane][kk + koff];
#pragma unroll
            for (int nt = 0; nt < 4; ++nt) {
                v2f b = *(const v2f*)&b_tile[buf][wave_n * 64 + nt * 16 + mlane][kk + koff];
                acc[nt] = __builtin_amdgcn_wmma_f32_16x16x4_f32(
                    false, a, false, b, (short)0, acc[nt], false, false);
            }
        }
        buf ^= 1;
    }

#pragma unroll
    for (int nt = 0; nt < 4; ++nt) {
        int col = n0 + wave_n * 64 + nt * 16 + mlane;
#pragma unroll
        for (int r = 0; r < 8; ++r) {
            int e  = m0 + wave_m * 16 + r + ((lane >> 4) << 3);
            int et = edge_types[e];
            int d  = dst[e];
            float v = acc[nt][r] + bias_table[et * 256 + col];
            atomicAdd(&agg[(size_t)d * 256 + col], v);
        }
    }
}

// ---------------------------------------------------------------------------
// Fused bidirectional GRU step.  gridDim = (B/16, 2 dirs), 256 threads.
// Computes gh = h_prev(16,256) @ Whh^T(768,256) with WMMA, stages gh in LDS,
// then applies GRU gate math using precomputed gi (= x@Wih^T + bih).
// dir 0: time t forward; dir 1: time S-1-t backward.  y written at stride 512.
// ---------------------------------------------------------------------------
__global__ __launch_bounds__(256) void gru_step_kernel(
    const float* __restrict__ gi_f, const float* __restrict__ gi_b,
    const float* __restrict__ Whh_f, const float* __restrict__ Whh_b,
    const float* __restrict__ bhh_f, const float* __restrict__ bhh_b,
    float* __restrict__ h_f, float* __restrict__ h_b,
    float* __restrict__ y_out, int t)
{
    __shared__ float a_tile[16][36];
    __shared__ float gh[16][776];

    const int dir   = blockIdx.y;
    const int tid   = threadIdx.x;
    const int lane  = tid & 31;
    const int wid   = tid >> 5;                 // 0..7, wave owns 96 cols
    const int b0    = blockIdx.x * 16;
    const int mlane = lane & 15;
    const int koff  = (lane >> 4) << 1;
    const int t_eff = dir ? (S_ - 1 - t) : t;

    const float* gi  = dir ? gi_b  : gi_f;
    const float* Whh = dir ? Whh_b : Whh_f;
    const float* bhh = dir ? bhh_b : bhh_f;
    float*       h   = dir ? h_b   : h_f;

    v8f acc[6] = {};
    const float* wrow[6];
#pragma unroll
    for (int nt = 0; nt < 6; ++nt) {
        int n = wid * 96 + nt * 16 + mlane;
        wrow[nt] = Whh + (size_t)n * 256;
    }

    const int r_ = tid >> 4;
    const int p_ = (tid & 15) << 1;

    for (int k0 = 0; k0 < 256; k0 += 32) {
        __syncthreads();
        *(v2f*)&a_tile[r_][p_] = *(const v2f*)(h + (size_t)(b0 + r_) * 256 + k0 + p_);
        __syncthreads();
#pragma unroll
        for (int kk = 0; kk < 32; kk += 4) {
            v2f a = *(const v2f*)&a_tile[mlane][kk + koff];
#pragma unroll
            for (int nt = 0; nt < 6; ++nt) {
                v2f b = *(const v2f*)(wrow[nt] + k0 + kk + koff);
                acc[nt] = __builtin_amdgcn_wmma_f32_16x16x4_f32(
                    false, a, false, b, (short)0, acc[nt], false, false);
            }
        }
    }

    // stage gh(16 x 768) into LDS
#pragma unroll
    for (int nt = 0; nt < 6; ++nt) {
        int col = wid * 96 + nt * 16 + mlane;
#pragma unroll
        for (int r = 0; r < 8; ++r) {
            int row = r + ((lane >> 4) << 3);
            gh[row][col] = acc[nt][r];
        }
    }
    __syncthreads();

    // gate math: thread -> row = tid>>4, 16 cols
    const int row = tid >> 4;
    const int jb  = tid & 15;
    const float* gir = gi + ((size_t)t_eff * B_ + b0 + row) * 768;
    float* hrow = h + (size_t)(b0 + row) * 256;
    float* yrow = y_out + ((size_t)t_eff * B_ + b0 + row) * 512 + dir * 256;

#pragma unroll
    for (int i = 0; i < 16; ++i) {
        int j = jb + (i << 4);
        float r = sigmoid_(gir[j]       + gh[row][j]       + bhh[j]);
        float z = sigmoid_(gir[256 + j] + gh[row][256 + j] + bhh[256 + j]);
        float n = tanhf   (gir[512 + j] + r * (gh[row][512 + j] + bhh[512 + j]));
        float hp = hrow[j];
        float hn = (1.0f - z) * n + z * hp;
        hrow[j] = hn;
        yrow[j] = hn;
    }
}

// ---------------------------------------------------------------------------
// Small elementwise / gather / pool kernels
// ---------------------------------------------------------------------------
__global__ void embed_kernel(const float* __restrict__ tab,
                             const int* __restrict__ types,
                             float* __restrict__ x)
{
    int i = blockIdx.x * 256 + threadIdx.x;      // T*256
    int n = i >> 8, c = i & 255;
    x[i] = tab[types[n] * 256 + c];
}

// node_bidir[n,:] = inp[(n%S)*B + n/S, :]   (512 wide)
__global__ void gather_bidir_kernel(const float* __restrict__ inp,
                                    float* __restrict__ outp)
{
    int i = blockIdx.x * 256 + threadIdx.x;      // T*512
    int n = i >> 9, c = i & 511;
    outp[i] = inp[(size_t)(((n & 63) << 8) + (n >> 6)) * 512 + c];
}

// bias_table[ne,n] = edge_table[ne,:] . Wmsg[n, 256:512] + bmsg[n]
__global__ void bias_table_kernel(const float* __restrict__ edge_table,
                                  const float* __restrict__ Wmsg,
                                  const float* __restrict__ bmsg,
                                  float* __restrict__ btab)
{
    int ne = blockIdx.x, n = threadIdx.x;
    const float* er = edge_table + ne * 256;
    const float* wr = Wmsg + (size_t)n * 512 + 256;
    float s = bmsg[n];
    for (int k = 0; k < 256; ++k) s = fmaf(er[k], wr[k], s);
    btab[ne * 256 + n] = s;
}

// GRU cell elementwise (bih/bhh already folded into gi/gh by the GEMM bias)
__global__ void cell_ew_kernel(const float* __restrict__ gi,
                               const float* __restrict__ gh,
                               float* __restrict__ h)
{
    int m = blockIdx.x, j = threadIdx.x;
    const float* gim = gi + (size_t)m * 768;
    const float* ghm = gh + (size_t)m * 768;
    float r = sigmoid_(gim[j]       + ghm[j]);
    float z = sigmoid_(gim[256 + j] + ghm[256 + j]);
    float n = tanhf   (gim[512 + j] + r * ghm[512 + j]);
    float hp = h[(size_t)m * 256 + j];
    h[(size_t)m * 256 + j] = (1.0f - z) * n + z * hp;
}

// e := sigmoid(g) * e
__global__ void gated_ew_kernel(const float* __restrict__ g, float* __restrict__ e)
{
    size_t i = (size_t)blockIdx.x * 256 + threadIdx.x;
    e[i] = sigmoid_(g[i]) * e[i];
}

// graph_emb[g,:512] = sum over 64 nodes of gated[g*64+pos, :512]
__global__ void pool_kernel(const float* __restrict__ gated, float* __restrict__ gemb)
{
    int g = blockIdx.x, j = threadIdx.x;
    float s0 = 0.0f, s1 = 0.0f;
    for (int pos = 0; pos < 64; ++pos) {
        const float* row = gated + ((size_t)(g * 64 + pos)) * 512;
        s0 += row[j];
        s1 += row[j + 256];
    }
    gemb[(size_t)g * 512 + j]       = s0;
    gemb[(size_t)g * 512 + 256 + j] = s1;
}

// cat[b,:1024] = [h_f[b,:256], h_b[b,:256], graph_emb[b,:512]]
__global__ void concat_kernel(const float* __restrict__ hf, const float* __restrict__ hb,
                              const float* __restrict__ gemb, float* __restrict__ outp)
{
    int i = blockIdx.x * 256 + threadIdx.x;      // B*1024
    int b = i >> 10, c = i & 1023;
    float v;
    if (c < 256)      v = hf[b * 256 + c];
    else if (c < 512) v = hb[b * 256 + (c - 256)];
    else              v = gemb[(size_t)b * 512 + (c - 512)];
    outp[i] = v;
}

// ---------------------------------------------------------------------------
// Host orchestration
// ---------------------------------------------------------------------------
static void launch_gemm(const float* A, int lda, const int* rowidx,
                        const float* W, int ldw, const float* bias,
                        float* C, int ldc, int M, int N, int K, hipStream_t s)
{
    dim3 grid(M / 64, N / 128);
    gemm_bias_kernel<<<grid, 256, 0, s>>>(A, lda, rowidx, W, ldw, bias, C, ldc, K);
}

extern "C" void kernel_launch(void* const* d_in, const int* in_sizes, int n_in,
                              void* d_out, int out_size, void* d_ws, size_t ws_size,
                              hipStream_t stream)
{
    (void)in_sizes; (void)n_in; (void)out_size; (void)ws_size;

    const float* node_table = (const float*)d_in[0];
    const float* Wih_l0     = (const float*)d_in[1];   // (2,768,256)
    const float* Wih_l12    = (const float*)d_in[2];   // (2,2,768,512)
    const float* Whh        = (const float*)d_in[3];   // (3,2,768,256)
    const float* bih        = (const float*)d_in[4];   // (3,2,768)
    const float* bhh        = (const float*)d_in[5];   // (3,2,768)
    const float* Wmd        = (const float*)d_in[6];   // (256,512)
    const float* bmd        = (const float*)d_in[7];
    const float* edge_table = (const float*)d_in[8];   // (16,256)
    const float* Wmsg       = (const float*)d_in[9];   // (256,512)
    const float* bmsg       = (const float*)d_in[10];
    const float* cell_Wih   = (const float*)d_in[11];  // (768,256)
    const float* cell_Whh   = (const float*)d_in[12];  // (768,256)
    const float* cell_bih   = (const float*)d_in[13];
    const float* cell_bhh   = (const float*)d_in[14];
    const float* Wgate      = (const float*)d_in[15];  // (512,256)
    const float* bgate      = (const float*)d_in[16];
    const float* Wemb       = (const float*)d_in[17];  // (512,256)
    const float* bemb       = (const float*)d_in[18];
    const float* Wglob      = (const float*)d_in[19];  // (256,1024)
    const float* bglob      = (const float*)d_in[20];
    const int* node_types   = (const int*)d_in[21];    // (S,B)
    const int* edge_types   = (const int*)d_in[22];    // (E,)
    const int* src          = (const int*)d_in[23];
    const int* dst          = (const int*)d_in[24];

    // Workspace layout (floats)
    float* f = (float*)d_ws;
    float* buf_in0 = f;                      f += (size_t)T_ * 512;
    float* buf_in1 = f;                      f += (size_t)T_ * 512;
    float* gi_f    = f;                      f += (size_t)T_ * 768;
    float* gi_b    = f;                      f += (size_t)T_ * 768;
    float* agg     = f;                      f += (size_t)T_ * 256;
    float* h_f     = f;                      f += (size_t)B_ * 256;
    float* h_b     = f;                      f += (size_t)B_ * 256;
    float* btab    = f;                      f += (size_t)NE_ * 256;
    float* gemb    = f;                      f += (size_t)B_ * 512;
    float* cat     = f;                      f += (size_t)B_ * 1024;

    float* node_h = (float*)d_out;                    // (T,256)
    float* merged = (float*)d_out + (size_t)T_ * 256; // (B,256)

    // 1) embedding lookup -> buf_in0 (T x 256, seq-major)
    embed_kernel<<<T_ * 256 / 256, 256, 0, stream>>>(node_table, node_types, buf_in0);

    // 2) 3 bidirectional GRU layers
    const float* cur = buf_in0;
    float* nxt = buf_in1;
    for (int l = 0; l < 3; ++l) {
        const int K = (l == 0) ? 256 : 512;
        const int lda = K;
        const float* Wf = (l == 0) ? Wih_l0
                                   : Wih_l12 + (size_t)(l - 1) * 2 * 768 * 512;
        const float* Wb = Wf + (size_t)768 * K;
        const float* bih_f = bih + (size_t)(l * 2 + 0) * 768;
        const float* bih_b = bih + (size_t)(l * 2 + 1) * 768;
        const float* bhh_f = bhh + (size_t)(l * 2 + 0) * 768;
        const float* bhh_b = bhh + (size_t)(l * 2 + 1) * 768;
        const float* Whh_f = Whh + (size_t)(l * 2 + 0) * 768 * 256;
        const float* Whh_b = Whh + (size_t)(l * 2 + 1) * 768 * 256;

        // gi = X @ Wih^T + bih, whole sequence at once, both directions
        launch_gemm(cur, lda, nullptr, Wf, K, bih_f, gi_f, 768, T_, 768, K, stream);
        launch_gemm(cur, lda, nullptr, Wb, K, bih_b, gi_b, 768, T_, 768, K, stream);

        (void)hipMemsetAsync(h_f, 0, (size_t)B_ * 256 * sizeof(float), stream);
        (void)hipMemsetAsync(h_b, 0, (size_t)B_ * 256 * sizeof(float), stream);

        for (int t = 0; t < S_; ++t) {
            gru_step_kernel<<<dim3(B_ / 16, 2), 256, 0, stream>>>(
                gi_f, gi_b, Whh_f, Whh_b, bhh_f, bhh_b, h_f, h_b, nxt, t);
        }
        const float* tmp = cur;
        cur = nxt;
        nxt = (float*)tmp;
        if (l == 0) nxt = buf_in0;   // layer0: in0 -> in1; afterwards ping-pong
    }
    // cur == buf_in1 (layer-2 output), h_f/h_b hold final hidden states

    // 3) permuted gather + node projection -> node_h (in d_out)
    gather_bidir_kernel<<<T_ * 512 / 256, 256, 0, stream>>>(cur, buf_in0);
    launch_gemm(buf_in0, 512, nullptr, Wmd, 512, bmd, node_h, 256, T_, 256, 512, stream);

    // 4) edge bias table (only NE=16 distinct edge embeddings)
    bias_table_kernel<<<NE_, 256, 0, stream>>>(edge_table, Wmsg, bmsg, btab);

    // 5) message passing, DEPTH rounds
    for (int dpt = 0; dpt < DEPTH_; ++dpt) {
        (void)hipMemsetAsync(agg, 0, (size_t)T_ * 256 * sizeof(float), stream);
        msg_gemm_scatter_kernel<<<dim3(E_ / 64, 256 / 128), 256, 0, stream>>>(
            node_h, src, Wmsg, 512, btab, edge_types, dst, agg, 256);
        launch_gemm(agg,    256, nullptr, cell_Wih, 256, cell_bih, gi_f, 768, T_, 768, 256, stream);
        launch_gemm(node_h, 256, nullptr, cell_Whh, 256, cell_bhh, gi_b, 768, T_, 768, 256, stream);
        cell_ew_kernel<<<T_, 256, 0, stream>>>(gi_f, gi_b, node_h);
    }

    // 6) gating + graph pooling
    launch_gemm(node_h, 256, nullptr, Wgate, 256, bgate, buf_in0, 512, T_, 512, 256, stream);
    launch_gemm(node_h, 256, nullptr, Wemb,  256, bemb,  buf_in1, 512, T_, 512, 256, stream);
    gated_ew_kernel<<<T_ * 512 / 256, 256, 0, stream>>>(buf_in0, buf_in1);
    pool_kernel<<<B_, 256, 0, stream>>>(buf_in1, gemb);

    // 7) final projection
    concat_kernel<<<B_ * 1024 / 256, 256, 0, stream>>>(h_f, h_b, gemb, cat);
    launch_gemm(cat, 1024, nullptr, Wglob, 1024, bglob, merged, 256, B_, 256, 1024, stream);
}